// RandomRectangleErasing_54065048322154
// MI455X (gfx1250) — compile-verified
//
#include <hip/hip_runtime.h>
#include <math.h>

// Image geometry (fixed by the reference).
#define IMG_B 64
#define IMG_C 3
#define IMG_H 512
#define IMG_W 512

typedef float v4f __attribute__((ext_vector_type(4)));

struct Rect { int px, py, w, h; };

// ---------------------------------------------------------------------------
// Kernel 1: per-batch rectangle parameters (64 lanes, trivial cost).
// Matches the JAX reference exactly:
//   area   = (0.02 + su*(0.33-0.02)) * H*W
//   ar_lo  = 0.3 + au*0.7 ; ar_hi = 1.0 + au*(3.3333-1.0)
//   aspect = round(cu) ? ar_lo : ar_hi          (jnp.round == RNE -> rintf)
//   h = clip(round(sqrt(area*aspect)), 1, H); w = clip(round(sqrt(area/aspect)), 1, W)
//   pos = floor(u * (dim - size + 1))
// ---------------------------------------------------------------------------
__global__ void erase_setup_kernel(const float* __restrict__ scale_u,
                                   const float* __restrict__ ar_u,
                                   const float* __restrict__ ar_choice_u,
                                   const float* __restrict__ px_u,
                                   const float* __restrict__ py_u,
                                   Rect* __restrict__ rects) {
    int i = threadIdx.x;
    if (i >= IMG_B) return;
    float area  = (0.02f + scale_u[i] * (0.33f - 0.02f)) * (float)(IMG_H * IMG_W);
    float ar_lo = 0.3f + ar_u[i] * (1.0f - 0.3f);
    float ar_hi = 1.0f + ar_u[i] * (3.3333f - 1.0f);
    float aspect = (rintf(ar_choice_u[i]) != 0.0f) ? ar_lo : ar_hi;
    float hf = fminf(fmaxf(rintf(sqrtf(area * aspect)), 1.0f), (float)IMG_H);
    float wf = fminf(fmaxf(rintf(sqrtf(area / aspect)), 1.0f), (float)IMG_W);
    int hh = (int)hf;
    int ww = (int)wf;
    Rect r;
    r.px = (int)floorf(px_u[i] * (float)(IMG_W - ww + 1));
    r.py = (int)floorf(py_u[i] * (float)(IMG_H - hh + 1));
    r.w  = ww;
    r.h  = hh;
    rects[i] = r;
}

// ---------------------------------------------------------------------------
// Kernel 2: streaming apply. One float4 per lane. A wave32 covers 32
// consecutive float4s = 128 consecutive floats inside ONE row of one (b,c)
// plane, so both the row test and the wave-x-span test are wave-uniform.
// Three uniform paths:
//   - wave disjoint from rectangle: async global->LDS->global DMA copy
//     (gfx1250 ASYNCcnt path, NT temporal hints, no VGPR data movement)
//   - wave fully inside rectangle:  NT store zeros, skip the read entirely
//   - boundary wave (<=2 per intersecting row): NT load, per-elem select
// ---------------------------------------------------------------------------
__global__ void __launch_bounds__(256)
erase_apply_kernel(const v4f* __restrict__ in,
                   v4f* __restrict__ out,
                   const Rect* __restrict__ rects) {
    __shared__ __align__(16) v4f lds_buf[256];

    const int tid = threadIdx.x;
    const int ib  = (int)blockIdx.x * 256 + tid;     // float4 index, < 2^24

    const int x0 = (ib & (IMG_W / 4 - 1)) << 2;      // first x of this float4
    const int yy = (ib >> 7) & (IMG_H - 1);          // row (uniform per wave)
    const int b  = (ib >> 16) / IMG_C;               // 2^16 float4 per plane

    const Rect r = rects[b];
    const int ylo = r.py, yhi = r.py + r.h;
    const int xlo = r.px, xhi = r.px + r.w;

    // Wave-uniform span: 32 aligned float4s never cross a row boundary.
    const int wb  = ib & ~31;                        // wave-base float4 index
    const int xw0 = (wb & (IMG_W / 4 - 1)) << 2;     // wave's first x
    const int xw1 = xw0 + 127;                       // wave's last x

    const bool inRow = (yy >= ylo) && (yy < yhi);

    if (!inRow || xw1 < xlo || xw0 >= xhi) {
        // Wave disjoint from the rectangle: pure streaming copy via the
        // CDNA5 async LDS DMA path, non-temporal in both directions.
        unsigned int lds = (unsigned int)(unsigned long long)&lds_buf[tid];
        const v4f* sp = in + ib;
        v4f* dp = out + ib;
        asm volatile("global_load_async_to_lds_b128 %0, %1, off th:TH_LOAD_NT"
                     :: "v"(lds), "v"(sp) : "memory");
        asm volatile("s_wait_asynccnt 0" ::: "memory");
        asm volatile("global_store_async_from_lds_b128 %0, %1, off th:TH_STORE_NT"
                     :: "v"(dp), "v"(lds) : "memory");
        // s_endpgm's implicit wait-idle covers the outstanding async store.
        return;
    }

    if (xw0 >= xlo && xw1 < xhi) {
        // Whole wave inside the rectangle: write zeros, skip the read.
        v4f z = {0.0f, 0.0f, 0.0f, 0.0f};
        __builtin_nontemporal_store(z, &out[ib]);
        return;
    }

    // Boundary wave: per-element select (rare — at most 2 waves per row).
    v4f v = __builtin_nontemporal_load(&in[ib]);
    v.x = (x0 + 0 >= xlo && x0 + 0 < xhi) ? 0.0f : v.x;
    v.y = (x0 + 1 >= xlo && x0 + 1 < xhi) ? 0.0f : v.y;
    v.z = (x0 + 2 >= xlo && x0 + 2 < xhi) ? 0.0f : v.z;
    v.w = (x0 + 3 >= xlo && x0 + 3 < xhi) ? 0.0f : v.w;
    __builtin_nontemporal_store(v, &out[ib]);
}

// ---------------------------------------------------------------------------
// Host launcher. Inputs (setup_inputs order):
//   0: images [64,3,512,512] f32   1: scale_u [64] f32   2: ar_u [64] f32
//   3: ar_choice_u [64] f32        4: px_u [64] f32      5: py_u [64] f32
// Output: [64,3,512,512] f32. Workspace: 64 * sizeof(Rect) = 1 KB.
// ---------------------------------------------------------------------------
extern "C" void kernel_launch(void* const* d_in, const int* in_sizes, int n_in,
                              void* d_out, int out_size, void* d_ws, size_t ws_size,
                              hipStream_t stream) {
    const float* images      = (const float*)d_in[0];
    const float* scale_u     = (const float*)d_in[1];
    const float* ar_u        = (const float*)d_in[2];
    const float* ar_choice_u = (const float*)d_in[3];
    const float* px_u        = (const float*)d_in[4];
    const float* py_u        = (const float*)d_in[5];

    Rect* rects = (Rect*)d_ws;

    erase_setup_kernel<<<1, 64, 0, stream>>>(scale_u, ar_u, ar_choice_u,
                                             px_u, py_u, rects);

    const int total_vec4 = IMG_B * IMG_C * IMG_H * IMG_W / 4;  // 12,582,912
    const int blocks = total_vec4 / 256;                        // 49,152
    erase_apply_kernel<<<blocks, 256, 0, stream>>>((const v4f*)images,
                                                   (v4f*)d_out, rects);
}